// RingModule_20005957665320
// MI455X (gfx1250) — compile-verified
//
#include <hip/hip_runtime.h>
#include <stdint.h>

// ---------------------------------------------------------------------------
// Ring-attractor RNN scan on MI455X (gfx1250):
//   h_{t+1} = 0.95*h_t + 0.05*relu(gamma*vel_t + bias + h_t @ W)
// T=8192 sequential steps, HIDDEN=2048, W = 2048x2048 f32 (16 MB).
//
// Latency-bound sequential scan -> 64 persistent WGs, W column-slices pinned
// in LDS via CDNA5 Tensor Data Mover (tensor_load_to_lds) with HW row padding
// (32+4 dwords) for conflict-free aligned ds_load_b128, device-wide counter
// barrier per step with agent release/acquire fences.
// ---------------------------------------------------------------------------

#define T_STEPS    8192
#define HIDDEN     2048
#define N_WG       64
#define COLS_WG    32          // HIDDEN / N_WG
#define THREADS    256         // 8 waves (wave32)
#define N_WAVE     8
#define COLS_WAVE  4
#define ROW_STRIDE 36          // 32 W cols + 4 pad dwords per LDS row (16B align)
#define SMEM_BYTES (ROW_STRIDE * HIDDEN * 4 + HIDDEN * 4)   // 294912 + 8192 B

typedef unsigned int v4u __attribute__((ext_vector_type(4)));
typedef int          v4i __attribute__((ext_vector_type(4)));
typedef int          v8i __attribute__((ext_vector_type(8)));

__global__ void zero_counter_kernel(unsigned* cnt) { *cnt = 0u; }

__device__ __forceinline__ void grid_sync(unsigned* cnt, unsigned target) {
  // publish this wave's global stores to device scope (s_wait_storecnt + wb)
  __builtin_amdgcn_fence(__ATOMIC_RELEASE, "agent");
  __syncthreads();
  if (threadIdx.x == 0) {
    __hip_atomic_fetch_add(cnt, 1u, __ATOMIC_RELAXED, __HIP_MEMORY_SCOPE_AGENT);
    while (__hip_atomic_load(cnt, __ATOMIC_RELAXED, __HIP_MEMORY_SCOPE_AGENT) < target) {
      __builtin_amdgcn_s_sleep(1);
    }
  }
  __syncthreads();
  // invalidate WGP$ so h loads observe other WGs' stores (global_inv)
  __builtin_amdgcn_fence(__ATOMIC_ACQUIRE, "agent");
}

__global__ void __launch_bounds__(THREADS, 1)
ring_scan_kernel(const float* __restrict__ x,
                 const float* __restrict__ h0,
                 const float* __restrict__ W,
                 const float* __restrict__ w_in,
                 const float* __restrict__ gamma,
                 const float* __restrict__ bias,
                 float* __restrict__ out,
                 unsigned* __restrict__ bar)
{
  extern __shared__ __align__(16) float smem[];
  float* Ws = smem;                          // padded W tile: ROW_STRIDE per row
  float* hs = smem + ROW_STRIDE * HIDDEN;    // h broadcast buffer (2048 f32)

  const int tid  = threadIdx.x;
  const int wave = tid >> 5;
  const int lane = tid & 31;
  const int j0   = blockIdx.x * COLS_WG;      // this WG's first column of W
  const int jw   = j0 + wave * COLS_WAVE;     // this wave's first column

  // ---- one-time: TDM DMA of W[:, j0:j0+32] (2048x32 f32) into padded LDS ---
#if __has_builtin(__builtin_amdgcn_tensor_load_to_lds)
  if (wave == 0) {
    const uint64_t ga   = (uint64_t)(uintptr_t)(W + j0);   // tile start address
    const unsigned lds0 = __builtin_amdgcn_groupstaticsize(); // dynamic-LDS base
    v4u g0;
    g0.x = 1u;                                                 // count=1 user D#
    g0.y = lds0;                                               // lds_addr = Ws
    g0.z = (unsigned)(ga & 0xFFFFFFFFu);                       // global_addr lo
    g0.w = (unsigned)((ga >> 32) & 0x01FFFFFFu) | (2u << 30);  // addr hi | type=2
    v8i g1;
    g1[0] = (int)((2u << 16)     // data_size = 4 bytes
                | (1u << 20)     // pad_enable (loads only)
                | (4u << 22)     // pad_interval: 2^(4+1) = 32 dwords (one row)
                | (3u << 25));   // pad_amount: 3+1 = 4 dwords -> row stride 36
    g1[1] = (int)((unsigned)HIDDEN << 16);   // tensor_dim0 = 2048 (lo16)
    g1[2] = (int)((unsigned)HIDDEN << 16);   // dim0 hi16=0 | tensor_dim1 lo16
    g1[3] = (int)((unsigned)COLS_WG << 16);  // dim1 hi16=0 | tile_dim0 = 32
    g1[4] = (int)HIDDEN;                     // tile_dim1 = 2048 | tile_dim2 = 0
    g1[5] = (int)HIDDEN;                     // tensor_dim0_stride = 2048 elems
    g1[6] = 0;                               // stride hi | tensor_dim1_stride
    g1[7] = 0;
    v4i gz = {0, 0, 0, 0};
#if defined(__clang_major__) && (__clang_major__ >= 23)
    v8i gz8 = {0, 0, 0, 0, 0, 0, 0, 0};
    __builtin_amdgcn_tensor_load_to_lds(g0, g1, gz, gz, gz8, 0);
#else
    __builtin_amdgcn_tensor_load_to_lds(g0, g1, gz, gz, 0);
#endif
    __builtin_amdgcn_s_wait_tensorcnt(0);
  }
#else
  // fallback: manual staged copy into the same padded layout
  for (int i = wave; i < HIDDEN; i += N_WAVE)
    Ws[i * ROW_STRIDE + lane] = W[(size_t)i * HIDDEN + j0 + lane];
#endif
  __syncthreads();

  // per-column constants (uniform within a wave -> scalar loads)
  float gv[COLS_WAVE], bv[COLS_WAVE];
#pragma unroll
  for (int c = 0; c < COLS_WAVE; ++c) {
    gv[c] = gamma[jw + c];
    bv[c] = bias[jw + c];
  }
  const float wi0 = w_in[0];
  const float wi1 = w_in[1];

  const float* hsrc = h0;
  float4 hkeep = make_float4(0.f, 0.f, 0.f, 0.f);
  const float* Wsw = Ws + wave * COLS_WAVE;   // this wave's 4-column base

  for (int t = 0; t < T_STEPS; ++t) {
    // stage h_t into LDS (coalesced b128 loads, conflict-free b128 LDS stores)
    {
      const float4* src4 = (const float4*)hsrc;
      float4*       dst4 = (float4*)hs;
      dst4[tid]           = src4[tid];
      dst4[tid + THREADS] = src4[tid + THREADS];
    }
    // issue the (uniform) input loads early: latency hides under the dot loop
    const float xv0 = x[2 * t];
    const float xv1 = x[2 * t + 1];
    __syncthreads();

    // 4-column register-blocked partial dot products, fully unrolled:
    // lane l accumulates i = l, l+32, ..., l+2016 (conflict-free b32 for h,
    // 16B-aligned conflict-free b128 for the 4 W columns)
    float a0 = 0.f, a1 = 0.f, a2 = 0.f, a3 = 0.f;
#pragma unroll
    for (int k = 0; k < HIDDEN / 32; ++k) {
      const int i = lane + 32 * k;
      const float hv = hs[i];
      const float4 wv = *(const float4*)&Wsw[i * ROW_STRIDE];
      a0 = __builtin_fmaf(hv, wv.x, a0);
      a1 = __builtin_fmaf(hv, wv.y, a1);
      a2 = __builtin_fmaf(hv, wv.z, a2);
      a3 = __builtin_fmaf(hv, wv.w, a3);
    }

    // issue h_prev load before the reduction so it hides under the shuffles
    const float4 hprev = *(const float4*)&hs[jw];

    // wave32 butterfly reduction (all lanes end with the column sums)
#pragma unroll
    for (int off = 16; off > 0; off >>= 1) {
      a0 += __shfl_xor(a0, off, 32);
      a1 += __shfl_xor(a1, off, 32);
      a2 += __shfl_xor(a2, off, 32);
      a3 += __shfl_xor(a3, off, 32);
    }

    if (lane == 0) {
      const float vel = wi0 * xv0 + wi1 * xv1;
      const float h2h[4] = {a0, a1, a2, a3};
      const float hp[4]  = {hprev.x, hprev.y, hprev.z, hprev.w};
      float4 hn;
      float* hnp = (float*)&hn;
#pragma unroll
      for (int c = 0; c < COLS_WAVE; ++c) {
        const float pre = __builtin_fmaf(gv[c], vel, bv[c]) + h2h[c];
        const float r   = pre > 0.f ? pre : 0.f;
        hnp[c] = __builtin_fmaf(0.95f, hp[c], 0.05f * r);
      }
      // output row t doubles as the h_{t+1} broadcast buffer
      *(float4*)(out + (size_t)t * HIDDEN + jw) = hn;
      hkeep = hn;
    }

    hsrc = out + (size_t)t * HIDDEN;
    if (t + 1 < T_STEPS) {
      grid_sync(bar, (unsigned)(N_WG * (t + 1)));
    }
  }

  // hidden_final (appended after hidden_acts) = last row, from registers
  if (lane == 0) {
    *(float4*)(out + (size_t)T_STEPS * HIDDEN + jw) = hkeep;
  }
}

extern "C" void kernel_launch(void* const* d_in, const int* in_sizes, int n_in,
                              void* d_out, int out_size, void* d_ws, size_t ws_size,
                              hipStream_t stream) {
  const float* x     = (const float*)d_in[0];  // (8192, 2)
  const float* h0    = (const float*)d_in[1];  // (2048,)
  const float* W     = (const float*)d_in[2];  // (2048, 2048)
  const float* w_in  = (const float*)d_in[3];  // (1, 2)
  const float* gamma = (const float*)d_in[4];  // (2048,)
  const float* bias  = (const float*)d_in[5];  // (2048,)
  float*    out = (float*)d_out;               // (8192*2048 + 2048,)
  unsigned* bar = (unsigned*)d_ws;

  (void)in_sizes; (void)n_in; (void)out_size; (void)ws_size;

  // allow >64KB dynamic LDS (303,104 B of the 320KB/WGP)
  hipFuncSetAttribute((const void*)ring_scan_kernel,
                      hipFuncAttributeMaxDynamicSharedMemorySize, SMEM_BYTES);

  zero_counter_kernel<<<1, 1, 0, stream>>>(bar);
  ring_scan_kernel<<<N_WG, THREADS, SMEM_BYTES, stream>>>(
      x, h0, W, w_in, gamma, bias, out, bar);
}